// GraphNet2_16080357556243
// MI455X (gfx1250) — compile-verified
//
#include <hip/hip_runtime.h>
#include <math.h>

#define N_NODES 10000
#define N_EDGES 320000
#define C_IN    128
#define H1      256
#define H2      256
#define EPSV    1e-5f

typedef float v2f __attribute__((ext_vector_type(2)));
typedef float v8f __attribute__((ext_vector_type(8)));

// ---------------------------------------------------------------------------
// BatchNorm: per-column stats (biased var), one block (256 thr) per column.
// ---------------------------------------------------------------------------
template<bool RELU>
__global__ void bn_stats_kernel(const float* __restrict__ X, int n, int c,
                                float* __restrict__ mean, float* __restrict__ inv) {
  __shared__ float sh_s[256];
  __shared__ float sh_q[256];
  const int col = blockIdx.x;
  float s = 0.f, q = 0.f;
  for (int i = threadIdx.x; i < n; i += 256) {
    float v = X[(size_t)i * c + col];
    if (RELU) v = fmaxf(v, 0.f);
    s += v;
    q += v * v;
  }
  sh_s[threadIdx.x] = s;
  sh_q[threadIdx.x] = q;
  __syncthreads();
  for (int st = 128; st > 0; st >>= 1) {
    if (threadIdx.x < st) {
      sh_s[threadIdx.x] += sh_s[threadIdx.x + st];
      sh_q[threadIdx.x] += sh_q[threadIdx.x + st];
    }
    __syncthreads();
  }
  if (threadIdx.x == 0) {
    float m   = sh_s[0] / (float)n;
    float var = sh_q[0] / (float)n - m * m;
    mean[col] = m;
    inv[col]  = rsqrtf(var + EPSV);
  }
}

template<bool RELU>
__global__ void bn_apply_kernel(const float* __restrict__ X,
                                const float* __restrict__ mean,
                                const float* __restrict__ inv,
                                const float* __restrict__ gamma,
                                const float* __restrict__ beta,
                                float* __restrict__ Y, int n, int c) {
  int idx = blockIdx.x * blockDim.x + threadIdx.x;
  if (idx >= n * c) return;
  int col = idx % c;
  float v = X[idx];
  if (RELU) v = fmaxf(v, 0.f);
  Y[idx] = (v - mean[col]) * inv[col] * gamma[col] + beta[col];
}

// ---------------------------------------------------------------------------
// CSR construction (deterministic aggregation, no float atomics)
// ---------------------------------------------------------------------------
__global__ void ifill_kernel(int* __restrict__ p, int v, int n) {
  int i = blockIdx.x * blockDim.x + threadIdx.x;
  if (i < n) p[i] = v;
}

__global__ void indeg_kernel(const int* __restrict__ dst, int e,
                             int* __restrict__ cnt) {
  int i = blockIdx.x * blockDim.x + threadIdx.x;
  if (i < e) atomicAdd(&cnt[dst[i]], 1);
}

// dis[i] = rsqrt(indegree + 1)  (self-loop included; always >= 1)
__global__ void dis_kernel(const int* __restrict__ cnt, float* __restrict__ dis,
                           int n) {
  int i = blockIdx.x * blockDim.x + threadIdx.x;
  if (i < n) dis[i] = rsqrtf((float)cnt[i] + 1.0f);
}

// single-block exclusive scan: rowptr[0..n] from cnt[0..n-1]
__global__ void exscan_kernel(const int* __restrict__ cnt,
                              int* __restrict__ rowptr, int n) {
  __shared__ int sh[256];
  __shared__ int s_carry;
  if (threadIdx.x == 0) s_carry = 0;
  for (int base = 0; base < n; base += 256) {
    __syncthreads();
    int cbase = s_carry;
    int i = base + threadIdx.x;
    int v = (i < n) ? cnt[i] : 0;
    sh[threadIdx.x] = v;
    __syncthreads();
    for (int off = 1; off < 256; off <<= 1) {
      int t = (threadIdx.x >= off) ? sh[threadIdx.x - off] : 0;
      __syncthreads();
      sh[threadIdx.x] += t;
      __syncthreads();
    }
    if (i < n) rowptr[i] = cbase + sh[threadIdx.x] - v;
    if (threadIdx.x == 0) s_carry = cbase + sh[255];
  }
  __syncthreads();
  if (threadIdx.x == 0) rowptr[n] = s_carry;
}

__global__ void csr_fill_kernel(const int* __restrict__ src,
                                const int* __restrict__ dst,
                                const int* __restrict__ rowptr,
                                int* __restrict__ cursor,
                                int* __restrict__ adj, int e) {
  int i = blockIdx.x * blockDim.x + threadIdx.x;
  if (i < e) {
    int d = dst[i];
    int pos = atomicAdd(&cursor[d], 1);
    adj[rowptr[d] + pos] = src[i];
  }
}

// per-node insertion sort of adjacency by src value: summation order becomes
// bitwise deterministic (ties are identical rows -> order irrelevant).
__global__ void adj_sort_kernel(const int* __restrict__ rowptr,
                                int* __restrict__ adj, int n) {
  int node = blockIdx.x * blockDim.x + threadIdx.x;
  if (node >= n) return;
  int beg = rowptr[node], end = rowptr[node + 1];
  for (int i = beg + 1; i < end; ++i) {
    int key = adj[i];
    int j = i - 1;
    while (j >= beg && adj[j] > key) { adj[j + 1] = adj[j]; --j; }
    adj[j + 1] = key;
  }
}

// ---------------------------------------------------------------------------
// Gather aggregations: one block of 64 threads per node, float4 per thread.
// Prefetch next neighbor row (emits global_prefetch_b8) to hide L2 latency.
// ---------------------------------------------------------------------------
__global__ __launch_bounds__(64)
void gcn_gather_kernel(const int* __restrict__ rowptr,
                       const int* __restrict__ adj,
                       const float* __restrict__ hlin,
                       const float* __restrict__ dis,
                       const float* __restrict__ bias,
                       float* __restrict__ out, int c) {
  const int node = blockIdx.x;
  const int q = threadIdx.x;                  // channels 4q..4q+3
  const int c4 = c >> 2;
  const float4* __restrict__ h4 = (const float4*)hlin;
  const int beg = rowptr[node];
  const int end = rowptr[node + 1];
  const float di = dis[node];
  const float4 b4 = ((const float4*)bias)[q];
  const float4 hv = h4[(size_t)node * c4 + q];
  const float w0 = di * di;
  float4 acc;
  acc.x = w0 * hv.x + b4.x;
  acc.y = w0 * hv.y + b4.y;
  acc.z = w0 * hv.z + b4.z;
  acc.w = w0 * hv.w + b4.w;
  for (int e = beg; e < end; ++e) {
    int s = adj[e];
    if (e + 1 < end)
      __builtin_prefetch(&h4[(size_t)adj[e + 1] * c4 + q], 0, 1);
    float w = di * dis[s];
    float4 v = h4[(size_t)s * c4 + q];
    acc.x += w * v.x;
    acc.y += w * v.y;
    acc.z += w * v.z;
    acc.w += w * v.w;
  }
  ((float4*)out)[(size_t)node * c4 + q] = acc;
}

__global__ __launch_bounds__(64)
void sum_gather_kernel(const int* __restrict__ rowptr,
                       const int* __restrict__ adj,
                       const float* __restrict__ h,
                       float* __restrict__ agg, int c) {
  const int node = blockIdx.x;
  const int q = threadIdx.x;
  const int c4 = c >> 2;
  const float4* __restrict__ h4 = (const float4*)h;
  const int beg = rowptr[node];
  const int end = rowptr[node + 1];
  float4 acc = make_float4(0.f, 0.f, 0.f, 0.f);
  for (int e = beg; e < end; ++e) {
    int s = adj[e];
    if (e + 1 < end)
      __builtin_prefetch(&h4[(size_t)adj[e + 1] * c4 + q], 0, 1);
    float4 v = h4[(size_t)s * c4 + q];
    acc.x += v.x;
    acc.y += v.y;
    acc.z += v.z;
    acc.w += v.w;
  }
  ((float4*)agg)[(size_t)node * c4 + q] = acc;
}

// ---------------------------------------------------------------------------
// FP32 WMMA GEMM: C[M,Nc] = A[M,K] @ B[K,Nc] (+ A2[M,K2] @ B2[K2,Nc]) (+bias)
// One 16x16 output tile per wave32, V_WMMA_F32_16X16X4_F32 K-steps of 4.
// ---------------------------------------------------------------------------
template<bool HAS2, bool BIAS, bool RELU>
__global__ __launch_bounds__(128)
void wmma_gemm_f32_kernel(const float* __restrict__ A, const float* __restrict__ B,
                          const float* __restrict__ A2, const float* __restrict__ B2,
                          const float* __restrict__ bias, float* __restrict__ C,
                          int M, int K, int K2, int Ncols) {
  const int wave  = threadIdx.x >> 5;
  const int tile  = blockIdx.x * 4 + wave;
  const int ntN   = Ncols >> 4;
  const int tm    = (tile / ntN) << 4;
  const int tn    = (tile % ntN) << 4;
  if (tm >= M) return;                    // uniform per-wave; exec stays full
  const int lane  = threadIdx.x & 31;
  const int half  = lane >> 4;
  const int l     = lane & 15;

  v8f acc;
  const float cinit = BIAS ? bias[tn + l] : 0.f;
#pragma unroll
  for (int r = 0; r < 8; ++r) acc[r] = cinit;

  {
    const float* arow = A + (size_t)(tm + l) * K;
    const float* bcol = B + tn + l;
    for (int k0 = 0; k0 < K; k0 += 4) {
      const int ka = k0 + 2 * half;
      v2f a, b;
      a.x = arow[ka];
      a.y = arow[ka + 1];
      b.x = bcol[(size_t)ka * Ncols];
      b.y = bcol[(size_t)(ka + 1) * Ncols];
      acc = __builtin_amdgcn_wmma_f32_16x16x4_f32(false, a, false, b,
                                                  (short)0, acc, false, false);
    }
  }
  if (HAS2) {
    const float* arow = A2 + (size_t)(tm + l) * K2;
    const float* bcol = B2 + tn + l;
    for (int k0 = 0; k0 < K2; k0 += 4) {
      const int ka = k0 + 2 * half;
      v2f a, b;
      a.x = arow[ka];
      a.y = arow[ka + 1];
      b.x = bcol[(size_t)ka * Ncols];
      b.y = bcol[(size_t)(ka + 1) * Ncols];
      acc = __builtin_amdgcn_wmma_f32_16x16x4_f32(false, a, false, b,
                                                  (short)0, acc, false, false);
    }
  }

#pragma unroll
  for (int r = 0; r < 8; ++r) {
    float v = acc[r];
    if (RELU) v = fmaxf(v, 0.f);
    C[(size_t)(tm + r + 8 * half) * Ncols + tn + l] = v;
  }
}

// ---------------------------------------------------------------------------
// Host-side launch sequence
// ---------------------------------------------------------------------------
extern "C" void kernel_launch(void* const* d_in, const int* in_sizes, int n_in,
                              void* d_out, int out_size, void* d_ws, size_t ws_size,
                              hipStream_t stream) {
  const float* x      = (const float*)d_in[0];
  const int*   ei     = (const int*)  d_in[1];      // [2, E] flat, int32
  const float* gamma0 = (const float*)d_in[2];
  const float* beta0  = (const float*)d_in[3];
  const float* Wg     = (const float*)d_in[4];      // [128, 256]
  const float* bg     = (const float*)d_in[5];
  const float* gamma1 = (const float*)d_in[6];
  const float* beta1  = (const float*)d_in[7];
  const float* Wrel   = (const float*)d_in[8];      // [256, 256]
  const float* brel   = (const float*)d_in[9];
  const float* Wroot  = (const float*)d_in[10];     // [256, 256]
  const float* gamma2 = (const float*)d_in[11];
  const float* beta2  = (const float*)d_in[12];
  float* out = (float*)d_out;

  const int* src = ei;
  const int* dst = ei + N_EDGES;

  // ---- workspace layout ----
  float* ws   = (float*)d_ws;
  float* xn   = ws;                              // N * 128
  float* hlin = xn   + (size_t)N_NODES * C_IN;   // N * 256 (reused: layer-2 out)
  float* gcn  = hlin + (size_t)N_NODES * H1;     // N * 256 (reused: agg)
  float* hbn  = gcn  + (size_t)N_NODES * H1;     // N * 256
  float* dis  = hbn  + (size_t)N_NODES * H2;     // N
  float* mean = dis  + N_NODES;                  // 256
  float* inv  = mean + 256;                      // 256
  int*   ip     = (int*)(inv + 256);
  int*   cnt    = ip;                            // N
  int*   cursor = cnt + N_NODES;                 // N
  int*   rowptr = cursor + N_NODES;              // N + 1
  int*   adj    = rowptr + N_NODES + 1;          // E

  const int TPB = 256;
  const int gridNC1 = (N_NODES * C_IN + TPB - 1) / TPB;
  const int gridNH  = (N_NODES * H1   + TPB - 1) / TPB;
  const int gridN   = (N_NODES + TPB - 1) / TPB;
  const int gridE   = (N_EDGES + TPB - 1) / TPB;
  const int gemmBlocks = (N_NODES / 16) * (H1 / 16) / 4;   // 4 waves per block

  // ---- CSR build (int atomics only; adjacency sorted -> deterministic) ----
  ifill_kernel<<<gridN, TPB, 0, stream>>>(cnt, 0, N_NODES);
  indeg_kernel<<<gridE, TPB, 0, stream>>>(dst, N_EDGES, cnt);
  dis_kernel<<<gridN, TPB, 0, stream>>>(cnt, dis, N_NODES);
  exscan_kernel<<<1, TPB, 0, stream>>>(cnt, rowptr, N_NODES);
  ifill_kernel<<<gridN, TPB, 0, stream>>>(cursor, 0, N_NODES);
  csr_fill_kernel<<<gridE, TPB, 0, stream>>>(src, dst, rowptr, cursor, adj, N_EDGES);
  adj_sort_kernel<<<gridN, TPB, 0, stream>>>(rowptr, adj, N_NODES);

  // ---- BN0 on x ----
  bn_stats_kernel<false><<<C_IN, TPB, 0, stream>>>(x, N_NODES, C_IN, mean, inv);
  bn_apply_kernel<false><<<gridNC1, TPB, 0, stream>>>(x, mean, inv, gamma0, beta0,
                                                      xn, N_NODES, C_IN);

  // ---- GCNConv: hlin = xn @ Wg  (WMMA f32) ----
  wmma_gemm_f32_kernel<false, false, false><<<gemmBlocks, 128, 0, stream>>>(
      xn, Wg, nullptr, nullptr, nullptr, hlin, N_NODES, C_IN, 0, H1);

  // gcn[i] = dis_i^2 * hlin[i] + bg + sum_s dis_i*dis_s*hlin[s]  (gather)
  gcn_gather_kernel<<<N_NODES, 64, 0, stream>>>(rowptr, adj, hlin, dis, bg,
                                                gcn, H1);

  // ---- BN1(relu(gcn)) -> hbn ----
  bn_stats_kernel<true><<<H1, TPB, 0, stream>>>(gcn, N_NODES, H1, mean, inv);
  bn_apply_kernel<true><<<gridNH, TPB, 0, stream>>>(gcn, mean, inv, gamma1, beta1,
                                                    hbn, N_NODES, H1);

  // ---- GraphConv: agg = gather_sum(hbn[adj]) (reuse gcn buffer) ----
  sum_gather_kernel<<<N_NODES, 64, 0, stream>>>(rowptr, adj, hbn, gcn, H1);

  // out2 = relu(agg @ Wrel + hbn @ Wroot + brel)  -> reuse hlin
  wmma_gemm_f32_kernel<true, true, true><<<gemmBlocks, 128, 0, stream>>>(
      gcn, Wrel, hbn, Wroot, brel, hlin, N_NODES, H1, H1, H2);

  // ---- BN2 -> d_out ----
  bn_stats_kernel<false><<<H2, TPB, 0, stream>>>(hlin, N_NODES, H2, mean, inv);
  bn_apply_kernel<false><<<gridNH, TPB, 0, stream>>>(hlin, mean, inv, gamma2, beta2,
                                                     out, N_NODES, H2);
}